// GCNLayer_9474697855478
// MI455X (gfx1250) — compile-verified
//
#include <hip/hip_runtime.h>
#include <stdint.h>

typedef __attribute__((ext_vector_type(2))) float v2f;
typedef __attribute__((ext_vector_type(8))) float v8f;
typedef __attribute__((ext_vector_type(4))) int   v4i;

// Address-space-qualified v4i for the async-copy builtin's typed pointers.
typedef __attribute__((address_space(1))) v4i g_v4i;  // global
typedef __attribute__((address_space(3))) v4i l_v4i;  // LDS

#define GCN_IN  512
#define GCN_OUT 256

// ---------------------------------------------------------------------------
// Async global->LDS copy (CDNA5 GLOBAL_LOAD_ASYNC_TO_LDS_B128, ASYNCcnt),
// with a synchronous register-copy fallback if the builtin is unavailable.
// ---------------------------------------------------------------------------
__device__ __forceinline__ void async_copy_b128(const void* g, void* l) {
#if defined(__AMDGCN__) && __has_builtin(__builtin_amdgcn_global_load_async_to_lds_b128)
  __builtin_amdgcn_global_load_async_to_lds_b128(
      (g_v4i*)g, (l_v4i*)l, /*imm offset*/ 0, /*cpol*/ 0);
#else
  *(float4*)l = *(const float4*)g;
#endif
}

__device__ __forceinline__ void wait_async_copies() {
#if defined(__AMDGCN__) && __has_builtin(__builtin_amdgcn_s_wait_asynccnt)
  __builtin_amdgcn_s_wait_asynccnt(0);
#endif
  // fallback path used synchronous loads; nothing to wait on
}

// ---------------------------------------------------------------------------
// Kernel 1: XW = x @ W   (M=N_nodes, K=512, N=256) via V_WMMA_F32_16X16X4_F32
// Block = 256 threads = 8 waves; block owns a 32-row strip of x (staged once
// into LDS, 64 KB, via async copies). Wave w computes cols [w*32, w*32+32)
// for all 32 rows -> 4 16x16 f32 accumulators (2 M-tiles x 2 N-tiles).
// Per K-step of 4: 2 ds_load_b64 (A) + 4 global b32 (B) + 4 WMMAs.
// ---------------------------------------------------------------------------
__global__ __launch_bounds__(256) void gcn_gemm_wmma(
    const float* __restrict__ x, const float* __restrict__ W,
    float* __restrict__ xw) {
  __shared__ float aTile[32 * GCN_IN];  // 64 KB: rows m0..m0+31 of x

  const int tid  = threadIdx.x;
  const int lane = tid & 31;
  const int wave = tid >> 5;           // 0..7
  const int m0   = blockIdx.x << 5;    // 32-row strip
  const int n0   = wave << 5;          // 32-col slab per wave
  const int half = lane >> 4;          // 0: K={k,k+1}, 1: K={k+2,k+3}
  const int l16  = lane & 15;

  // ---- Stage contiguous 64 KB strip of x into LDS (16 b128 per thread) ----
  {
    const char* gsrc = (const char*)(x + (size_t)m0 * GCN_IN);
    char*       ldst = (char*)aTile;
#pragma unroll
    for (int p = 0; p < 16; ++p) {
      const int off = (tid << 4) + (p << 12);  // tid*16 + p*4096
      async_copy_b128(gsrc + off, ldst + off);
    }
  }
  wait_async_copies();
  __syncthreads();

  // A fragment bases in LDS (float offsets): row-major 32 x 512.
  const float* a0Base = aTile + (size_t)l16 * GCN_IN + (half << 1);         // rows 0..15
  const float* a1Base = aTile + (size_t)(16 + l16) * GCN_IN + (half << 1);  // rows 16..31
  // B fragment bases for the two 16-wide N tiles (row-major W: stride 256).
  const float* bPtr0 = W + (size_t)(half << 1) * GCN_OUT + n0 + l16;
  const float* bPtr1 = bPtr0 + 16;

  v8f c00 = {}, c01 = {}, c10 = {}, c11 = {};

#pragma unroll 4
  for (int k = 0; k < GCN_IN; k += 4) {
    v2f a0 = *(const v2f*)(a0Base + k);  // ds_load_b64
    v2f a1 = *(const v2f*)(a1Base + k);  // ds_load_b64
    v2f b0, b1;
    const size_t koff = (size_t)k * GCN_OUT;
    b0.x = bPtr0[koff];
    b0.y = bPtr0[koff + GCN_OUT];
    b1.x = bPtr1[koff];
    b1.y = bPtr1[koff + GCN_OUT];
    c00 = __builtin_amdgcn_wmma_f32_16x16x4_f32(false, a0, false, b0,
                                                (short)0, c00, false, false);
    c01 = __builtin_amdgcn_wmma_f32_16x16x4_f32(false, a0, false, b1,
                                                (short)0, c01, false, false);
    c10 = __builtin_amdgcn_wmma_f32_16x16x4_f32(false, a1, false, b0,
                                                (short)0, c10, false, false);
    c11 = __builtin_amdgcn_wmma_f32_16x16x4_f32(false, a1, false, b1,
                                                (short)0, c11, false, false);
  }

  // C/D layout: VGPR r -> M = r + half*8, N = l16.
  float* o = xw + (size_t)m0 * GCN_OUT + n0 + l16;
#pragma unroll
  for (int r = 0; r < 8; ++r) {
    const size_t row0 = (size_t)(r + (half << 3)) * GCN_OUT;
    const size_t row1 = row0 + (size_t)16 * GCN_OUT;
    o[row0]      = c00[r];
    o[row0 + 16] = c01[r];
    o[row1]      = c10[r];
    o[row1 + 16] = c11[r];
  }
}

// ---------------------------------------------------------------------------
// Kernel 2: out[n][c] = b[c]   (bias pre-seeded so the scatter accumulates
// on top; float4-vectorized, total = N*64 float4s)
// ---------------------------------------------------------------------------
__global__ __launch_bounds__(256) void gcn_bias_init(
    float* __restrict__ out, const float* __restrict__ bias, int total4) {
  int i = blockIdx.x * blockDim.x + threadIdx.x;
  if (i >= total4) return;
  const int c = (i & 63) << 2;
  float4 bb = *(const float4*)(bias + c);
  ((float4*)out)[i] = bb;
}

// ---------------------------------------------------------------------------
// Kernel 3: COO scatter  out[row] += val * xw[col]
// One wave32 per edge; lane handles 8 consecutive columns (2 b128 gathers
// from L2-resident xw + 8 global_atomic_add_f32).
// ---------------------------------------------------------------------------
__global__ __launch_bounds__(256) void gcn_edge_scatter(
    const int* __restrict__ erow, const int* __restrict__ ecol,
    const float* __restrict__ eval, const float* __restrict__ xw,
    float* __restrict__ out, int E) {
  const int wavesPerBlock = blockDim.x >> 5;
  const int e = blockIdx.x * wavesPerBlock + (threadIdx.x >> 5);
  if (e >= E) return;
  const int lane = threadIdx.x & 31;

  const int   row = erow[e];
  const int   col = ecol[e];
  const float v   = eval[e];

  const float* s = xw  + (size_t)col * GCN_OUT + (lane << 3);
  float*       d = out + (size_t)row * GCN_OUT + (lane << 3);

  float4 s0 = *(const float4*)(s);
  float4 s1 = *(const float4*)(s + 4);

  atomicAdd(d + 0, v * s0.x);
  atomicAdd(d + 1, v * s0.y);
  atomicAdd(d + 2, v * s0.z);
  atomicAdd(d + 3, v * s0.w);
  atomicAdd(d + 4, v * s1.x);
  atomicAdd(d + 5, v * s1.y);
  atomicAdd(d + 6, v * s1.z);
  atomicAdd(d + 7, v * s1.w);
}

// ---------------------------------------------------------------------------
// Kernel 4: in-place ReLU on out (float4-vectorized)
// ---------------------------------------------------------------------------
__global__ __launch_bounds__(256) void gcn_relu(float* __restrict__ out,
                                                int total4) {
  int i = blockIdx.x * blockDim.x + threadIdx.x;
  if (i >= total4) return;
  float4 v = ((float4*)out)[i];
  v.x = fmaxf(v.x, 0.0f);
  v.y = fmaxf(v.y, 0.0f);
  v.z = fmaxf(v.z, 0.0f);
  v.w = fmaxf(v.w, 0.0f);
  ((float4*)out)[i] = v;
}

extern "C" void kernel_launch(void* const* d_in, const int* in_sizes, int n_in,
                              void* d_out, int out_size, void* d_ws,
                              size_t ws_size, hipStream_t stream) {
  const float* x    = (const float*)d_in[0];
  const float* W    = (const float*)d_in[1];
  const float* bias = (const float*)d_in[2];
  const int*   erow = (const int*)d_in[3];
  const int*   ecol = (const int*)d_in[4];
  const float* ev   = (const float*)d_in[5];
  float*       out  = (float*)d_out;
  float*       xw   = (float*)d_ws;  // needs N*256*4 = 102.4 MB

  const int N = in_sizes[0] / GCN_IN;  // 100000
  const int E = in_sizes[3];           // 3200000

  // 1) XW = x @ W  (100000 = 3125 * 32)
  gcn_gemm_wmma<<<N / 32, 256, 0, stream>>>(x, W, xw);

  // 2) out = b (broadcast rows)
  const int total4 = N * (GCN_OUT / 4);
  gcn_bias_init<<<(total4 + 255) / 256, 256, 0, stream>>>(out, bias, total4);

  // 3) out[row] += val * xw[col] over all edges (1 wave per edge)
  gcn_edge_scatter<<<(E + 7) / 8, 256, 0, stream>>>(erow, ecol, ev, xw, out, E);

  // 4) out = relu(out)
  gcn_relu<<<(total4 + 255) / 256, 256, 0, stream>>>(out, total4);
}